// RepresentationEncoderState_1133871366358
// MI455X (gfx1250) — compile-verified
//
#include <hip/hip_runtime.h>
#include <cstddef>
#include <cstdint>

typedef __attribute__((ext_vector_type(2))) float v2f;
typedef __attribute__((ext_vector_type(8))) float v8f;
typedef __attribute__((ext_vector_type(4))) unsigned int v4u;
typedef __attribute__((ext_vector_type(8))) int v8i;
typedef __attribute__((ext_vector_type(4))) int v4i;

#define ZONEOUT 0.15f
#define BATCH 512
#define TSTEPS 256
#define DIN 256
#define HID 128
#define FOURH 512   // 4*HID

__device__ __forceinline__ float sigm(float v) {
    return 1.0f / (1.0f + __expf(-v));
}
__device__ __forceinline__ float tanh_fast(float v) {
    float e = __expf(2.0f * v);
    return (e - 1.0f) / (e + 1.0f);
}

__device__ __forceinline__ v8f wmma_f32(v2f a, v2f b, v8f c) {
    // D = A(16x4) x B(4x16) + C(16x16), fp32
    return __builtin_amdgcn_wmma_f32_16x16x4_f32(
        /*neg_a=*/false, a, /*neg_b=*/false, b,
        /*c_mod=*/(short)0, c, /*reuse_a=*/false, /*reuse_b=*/false);
}

// ---------------------------------------------------------------------------
// TDM: DMA a [rows x 256] fp32 tile (row-major, stride 256) into LDS.
// Descriptor layout per CDNA5 ISA (D# groups 0/1).
// clang-23 toolchain: 6-arg builtin (g0 v4u, g1 v8i, g2 v4i, g3 v4i, v8i, cpol)
// ---------------------------------------------------------------------------
__device__ __forceinline__ void tdm_load_x_tile(const float* gsrc, void* lds_dst,
                                                unsigned int rows) {
    const unsigned long long ga = (unsigned long long)(uintptr_t)gsrc;
    const unsigned int lds_addr = (unsigned int)(uintptr_t)lds_dst;

    v4u g0;
    g0.x = 1u;                                   // count=1 (valid), user mode
    g0.y = lds_addr;                             // lds_addr [63:32]
    g0.z = (unsigned int)ga;                     // global_addr [95:64]
    g0.w = (unsigned int)(ga >> 32) | 0x80000000u; // global_addr hi + type=2 @[127:126]

    v8i g1;
    g1[0] = 0x20000;                             // data_size=2 (4 bytes), no flags
    g1[1] = (int)((DIN & 0xFFFF) << 16);         // tensor_dim0 = 256 (lo16 in [31:16])
    g1[2] = (int)((DIN >> 16) | ((rows & 0xFFFF) << 16)); // dim0 hi | tensor_dim1 lo
    g1[3] = (int)((rows >> 16) | (DIN << 16));   // tensor_dim1 hi | tile_dim0 = 256
    g1[4] = (int)rows;                           // tile_dim1 = rows, tile_dim2 = 0
    g1[5] = DIN;                                 // tensor_dim0_stride = 256 (lo32)
    g1[6] = (int)((DIN & 0xFFFF) << 16);         // stride0 hi16 | tensor_dim1_stride lo16
    g1[7] = 0;                                   // stride1 hi

    v4i gz4 = {0, 0, 0, 0};
    v8i gz8 = {0, 0, 0, 0, 0, 0, 0, 0};
    __builtin_amdgcn_tensor_load_to_lds(g0, g1, gz4, gz4, gz8, /*cpol=*/0);
    __builtin_amdgcn_s_wait_tensorcnt(0);
}

// ---------------------------------------------------------------------------
// Kernel A: P[B*T, 4H] = x[B*T, D] @ Wx^T + b   (time-parallel, big GEMM)
// Block tile: 64 rows x 512 cols. 8 waves; wave w -> N-tiles {4w..4w+3},
// each over 4 M-tiles (C = 16 fragments = 128 VGPRs). x tile (64x256, 64 KB)
// DMA'd to LDS by the Tensor Data Mover; Wx B-fragments stream from L2
// (512 KB per block -> ~1 GB total L2 traffic, 4x less than a 16-row tile).
// ---------------------------------------------------------------------------
__global__ __launch_bounds__(256) void xproj_kernel(
    const float* __restrict__ x, const float* __restrict__ Wx,
    const float* __restrict__ bias, float* __restrict__ P)
{
    __shared__ float xs[64 * DIN];    // 64 KB

    const int tid  = threadIdx.x;
    const int lane = tid & 31;
    const int wave = tid >> 5;        // 0..7
    const int l15  = lane & 15;
    const int khi  = lane >> 4;       // 0: K={0,1}, 1: K={2,3}
    const size_t row0 = (size_t)blockIdx.x * 64;

    // Async DMA of the x tile [64][256] into LDS (one TDM op from wave 0)
    if (wave == 0) {
        tdm_load_x_tile(x + row0 * DIN, xs, 64u);
    }
    __syncthreads();

    const int n0 = wave * 64;         // this wave's first column
    v8f c[4][4];                      // [m-tile][n-tile]
    #pragma unroll
    for (int mt = 0; mt < 4; ++mt)
        #pragma unroll
        for (int nt = 0; nt < 4; ++nt) c[mt][nt] = v8f{};

    #pragma unroll 2
    for (int k = 0; k < DIN; k += 4) {
        // A fragments for the 4 M-tiles (8B-aligned LDS reads)
        v2f a[4];
        #pragma unroll
        for (int mt = 0; mt < 4; ++mt)
            a[mt] = *(const v2f*)(&xs[(mt * 16 + l15) * DIN + k + 2 * khi]);
        #pragma unroll
        for (int nt = 0; nt < 4; ++nt) {
            const int n = n0 + nt * 16 + l15;
            // B fragment: Wx[n][k + 2*khi .. +1] (Wx row-major over K)
            v2f bb = *(const v2f*)(&Wx[(size_t)n * DIN + k + 2 * khi]);
            #pragma unroll
            for (int mt = 0; mt < 4; ++mt)
                c[mt][nt] = wmma_f32(a[mt], bb, c[mt][nt]);
        }
    }

    // Add bias (one scalar per lane: C lane -> fixed N) and store P
    #pragma unroll
    for (int nt = 0; nt < 4; ++nt) {
        const int n  = n0 + nt * 16 + l15;
        const float bn = bias[n];
        #pragma unroll
        for (int mt = 0; mt < 4; ++mt) {
            float* pout = P + (row0 + mt * 16) * FOURH + n;
            #pragma unroll
            for (int m = 0; m < 8; ++m) {
                const int M = m + 8 * khi;    // C: lanes>=16 hold M=8..15
                pout[(size_t)M * FOURH] = c[mt][nt][m] + bn;
            }
        }
    }
}

// ---------------------------------------------------------------------------
// Kernel B: sequential scan. One block = 16 batch rows, 512 threads = 16 waves.
// Wave w owns gate columns [32w, 32w+32): 2 WMMA N-tiles, K=128 (32 k-steps).
// Wh B-fragments are loop-invariant -> pre-loaded to registers (128 VGPR/wave).
// hid tile in LDS feeds A-fragments; cel/pog live in per-thread registers.
// ---------------------------------------------------------------------------
__global__ __launch_bounds__(512) void lstm_scan_kernel(
    const float* __restrict__ P, const float* __restrict__ Wh,
    float* __restrict__ hids, float* __restrict__ cels,
    float* __restrict__ pogs)
{
    __shared__ float lds_hid[16 * HID];      // 8 KB
    __shared__ float lds_gates[16 * FOURH];  // 32 KB

    const int tid  = threadIdx.x;
    const int lane = tid & 31;
    const int wave = tid >> 5;    // 0..15
    const int l15  = lane & 15;
    const int khi  = lane >> 4;
    const int b0   = blockIdx.x * 16;

    // ---- preload Wh B-fragments (loop-invariant over t) ----
    v2f bfrag[2][32];
    #pragma unroll
    for (int tI = 0; tI < 2; ++tI) {
        const int n = wave * 32 + tI * 16 + l15;
        const float* wrow = Wh + (size_t)n * HID;
        #pragma unroll
        for (int ks = 0; ks < 32; ++ks)
            bfrag[tI][ks] = *(const v2f*)(&wrow[ks * 4 + 2 * khi]);
    }

    // zero initial hidden state
    for (int i = tid; i < 16 * HID; i += 512) lds_hid[i] = 0.0f;

    // elementwise mapping: 512 threads x 4 elems = 16 rows x 128 h
    const int erow = tid >> 5;             // row 0..15
    const int h0   = (tid & 31) * 4;       // h offset, float4 aligned
    float cel[4] = {0, 0, 0, 0};
    float pog[4] = {0, 0, 0, 0};

    __syncthreads();

    for (int t = 0; t < TSTEPS; ++t) {
        // ---- gates_wmma = hid @ Wh^T ----
        v8f c0 = v8f{}, c1 = v8f{};
        #pragma unroll
        for (int ks = 0; ks < 32; ++ks) {
            v2f a = *(const v2f*)(&lds_hid[l15 * HID + ks * 4 + 2 * khi]);
            c0 = wmma_f32(a, bfrag[0][ks], c0);
            c1 = wmma_f32(a, bfrag[1][ks], c1);
        }
        // scatter C fragments to LDS gate buffer [16][512]
        #pragma unroll
        for (int m = 0; m < 8; ++m) {
            const int M = m + 8 * khi;
            lds_gates[M * FOURH + wave * 32 + l15]      = c0[m];
            lds_gates[M * FOURH + wave * 32 + 16 + l15] = c1[m];
        }
        __syncthreads();

        // ---- elementwise: gates = P[b,t] + gates_wmma; LSTM cell ----
        const float* Prow = P + ((size_t)(b0 + erow) * TSTEPS + t) * FOURH;
        const float* grow = lds_gates + erow * FOURH;

        float4 iP = *(const float4*)(Prow + 0 * HID + h0);
        float4 fP = *(const float4*)(Prow + 1 * HID + h0);
        float4 oP = *(const float4*)(Prow + 2 * HID + h0);
        float4 gP = *(const float4*)(Prow + 3 * HID + h0);
        float4 iW = *(const float4*)(grow + 0 * HID + h0);
        float4 fW = *(const float4*)(grow + 1 * HID + h0);
        float4 oW = *(const float4*)(grow + 2 * HID + h0);
        float4 gW = *(const float4*)(grow + 3 * HID + h0);

        const float* ip = (const float*)&iP; const float* iw = (const float*)&iW;
        const float* fp = (const float*)&fP; const float* fw = (const float*)&fW;
        const float* op = (const float*)&oP; const float* ow = (const float*)&oW;
        const float* gp = (const float*)&gP; const float* gw = (const float*)&gW;

        float4 hn4, cn4, og4;
        float* hn = (float*)&hn4; float* cn = (float*)&cn4; float* og = (float*)&og4;
        #pragma unroll
        for (int j = 0; j < 4; ++j) {
            const float gi = sigm(ip[j] + iw[j]);
            const float gf = sigm(fp[j] + fw[j]);
            const float go = (1.0f - ZONEOUT) * sigm(op[j] + ow[j]) + ZONEOUT * pog[j];
            const float gg = tanh_fast(gp[j] + gw[j]);
            const float cc = gf * cel[j] + gi * gg;
            cel[j] = cc;
            pog[j] = go;
            cn[j]  = cc;
            og[j]  = go;
            hn[j]  = go * tanh_fast(cc);
        }

        // new hidden -> LDS for next step's A-fragments
        *(float4*)(&lds_hid[erow * HID + h0]) = hn4;

        // outputs [B, T, H]
        const size_t obase = ((size_t)(b0 + erow) * TSTEPS + t) * HID + h0;
        *(float4*)(hids + obase) = hn4;
        *(float4*)(cels + obase) = cn4;
        *(float4*)(pogs + obase) = og4;

        __syncthreads();   // protects lds_hid and lds_gates for next iter
    }
}

extern "C" void kernel_launch(void* const* d_in, const int* in_sizes, int n_in,
                              void* d_out, int out_size, void* d_ws, size_t ws_size,
                              hipStream_t stream) {
    (void)in_sizes; (void)n_in; (void)out_size; (void)ws_size;
    const float* x   = (const float*)d_in[0];   // [B, T, D]
    const float* Wx  = (const float*)d_in[1];   // [4H, D]
    const float* Wh  = (const float*)d_in[2];   // [4H, H]
    const float* bia = (const float*)d_in[3];   // [4H]

    float* P = (float*)d_ws;                    // [B*T, 4H] = 256 MB scratch

    float* out  = (float*)d_out;
    const size_t plane = (size_t)BATCH * TSTEPS * HID;
    float* hids = out;
    float* cels = out + plane;
    float* pogs = out + 2 * plane;

    // 1) time-parallel input projection: 2048 blocks x 256 threads (64-row tiles)
    xproj_kernel<<<dim3((BATCH * TSTEPS) / 64), dim3(256), 0, stream>>>(x, Wx, bia, P);

    // 2) batch-parallel sequential scan: 32 blocks x 512 threads
    lstm_scan_kernel<<<dim3(BATCH / 16), dim3(512), 0, stream>>>(P, Wh, hids, cels, pogs);
}